// DipAwareLoss_73796128080203
// MI455X (gfx1250) — compile-verified
//
#include <hip/hip_runtime.h>

typedef __attribute__((ext_vector_type(2))) float v2f;
typedef __attribute__((ext_vector_type(8))) float v8f;

#define LNUM 1001
#define WINN 21
#define ROI_LO_I 140   // lam = 250 + 0.5*i ; 320 -> 140, 500 -> 500 (exact in f32)
#define ROI_HI_I 500

// One block = one spectrum row. 256 threads = 8 wave32.
__global__ __launch_bounds__(256) void dip_main(const float* __restrict__ pred,
                                                const float* __restrict__ tgt,
                                                const float* __restrict__ lam,
                                                float* __restrict__ gacc)
{
    __shared__ float tPad[1040];     // zero-padded target row (pad 6 each side + tile slack)
    __shared__ float sc[1004];       // ROI-masked -LoG scores
    __shared__ float msk[1004];      // keep-masked scores for top-k
    __shared__ float Wm[32][16];     // banded Toeplitz kernel matrix (K padded 28->32)
    __shared__ float kerS[16];
    __shared__ float rv[256];
    __shared__ int   ri[256];
    __shared__ int   sCen[6];
    __shared__ float sVal[6];
    __shared__ float lacc[4];

    const int tid = threadIdx.x;
    const int b   = blockIdx.x;
    const float* trow = tgt  + (size_t)b * LNUM;
    const float* prow = pred + (size_t)b * LNUM;

    __builtin_prefetch(prow, 0, 0);  // global_prefetch_b8: warm pred row for the dip stage

    // Load target row with zero padding (conv 'same' zero pad semantics).
    for (int i = tid; i < 1040; i += 256)
        tPad[i] = (i >= 6 && i < 6 + LNUM) ? trow[i - 6] : 0.0f;
    if (tid < 4) lacc[tid] = 0.0f;

    // LoG taps: sigma = max(1, 1.0/0.5) = 2, radius 6, mean-subtracted (symmetric -> no flip issue).
    if (tid == 0) {
        float raw[13]; float s = 0.0f;
        for (int t = 0; t < 13; ++t) {
            float x = (float)(t - 6);
            float g = expf(-x * x * (1.0f / 8.0f));
            raw[t] = (x * x - 4.0f) * (1.0f / 16.0f) * g;
            s += raw[t];
        }
        s *= (1.0f / 13.0f);
        for (int t = 0; t < 13; ++t) kerS[t] = raw[t] - s;
    }
    __syncthreads();

    // Banded Toeplitz W[c][j] = ker[c-j] for 0 <= c-j <= 12, else 0. (32x16, K-padded)
    for (int idx = tid; idx < 512; idx += 256) {
        int c = idx >> 4, j = idx & 15, d = c - j;
        Wm[c][j] = (d >= 0 && d <= 12) ? kerS[d] : 0.0f;
    }
    __syncthreads();

    const int wave = tid >> 5, lane = tid & 31;
    const int lo = lane & 15, hi = lane >> 4;

    // ---- conv-as-GEMM via V_WMMA_F32_16X16X4_F32 (full f32 precision) ----
    // Tile of 256 output positions per wave: S[m,j] = sum_c X[m,c]*W[c,j],
    // X[m,c] = tPad[base + 16m + c]  (score[p] = sum_t ker[t]*x[p+t-6], c = j+t).
    if (wave < 4) {
        const int base = wave * 256;
        v2f Bm[8];
#pragma unroll
        for (int kk = 0; kk < 8; ++kk) {     // B 4x16: lanes0-15 rows K=0/1, lanes16-31 rows K=2/3
            int c0 = 4 * kk + 2 * hi;
            Bm[kk].x = Wm[c0][lo];
            Bm[kk].y = Wm[c0 + 1][lo];
        }
        v8f d = {};
#pragma unroll
        for (int kk = 0; kk < 8; ++kk) {     // A 16x4: lane m=lo, K pair = 2*hi
            int c0 = 4 * kk + 2 * hi;
            v2f a;
            a.x = tPad[base + 16 * lo + c0];
            a.y = tPad[base + 16 * lo + c0 + 1];
            d = __builtin_amdgcn_wmma_f32_16x16x4_f32(false, a, false, Bm[kk],
                                                      (short)0, d, false, false);
        }
#pragma unroll
        for (int r = 0; r < 8; ++r) {        // D: M = 8*hi + r, N = lo
            int p = base + 128 * hi + 16 * r + lo;
            if (p < LNUM)
                sc[p] = (p >= ROI_LO_I && p <= ROI_HI_I) ? -d[r] : 0.0f;
        }
    }
    __syncthreads();

    // ---- row mean of masked scores ----
    float ls = 0.0f;
    for (int p = tid; p < LNUM; p += 256) ls += sc[p];
    rv[tid] = ls; __syncthreads();
    for (int s = 128; s > 0; s >>= 1) { if (tid < s) rv[tid] += rv[tid + s]; __syncthreads(); }
    const float meanv = rv[0] / 1001.0f;
    __syncthreads();

    // ---- local-max keep (pool width 11, -inf edge pad) ----
    const float NINF = -__builtin_inff();
    for (int p = tid; p < LNUM; p += 256) {
        float sp = sc[p];
        int q0 = p - 5 < 0 ? 0 : p - 5;
        int q1 = p + 5 > LNUM - 1 ? LNUM - 1 : p + 5;
        float mx = NINF;
        for (int q = q0; q <= q1; ++q) mx = fmaxf(mx, sc[q]);
        msk[p] = (sp == mx && sp > meanv) ? sp : NINF;
    }
    __syncthreads();

    // ---- top-6 (ties -> lowest index, matching lax.top_k stability) ----
    for (int k = 0; k < 6; ++k) {
        float bv = NINF; int bi = LNUM;
        for (int p = tid; p < LNUM; p += 256) {
            float v = msk[p];
            if (v > bv || (v == bv && p < bi)) { bv = v; bi = p; }
        }
        rv[tid] = bv; ri[tid] = bi; __syncthreads();
        for (int s = 128; s > 0; s >>= 1) {
            if (tid < s) {
                float ov = rv[tid + s]; int oi = ri[tid + s];
                if (ov > rv[tid] || (ov == rv[tid] && oi < ri[tid])) { rv[tid] = ov; ri[tid] = oi; }
            }
            __syncthreads();
        }
        if (tid == 0) {
            sVal[k] = rv[0];
            int c = ri[0] < LNUM ? ri[0] : 0;
            sCen[k] = c;
            msk[c] = NINF;
        }
        __syncthreads();
    }

    // ---- per-dip metrics: 6 waves, one dip each, 21 active lanes ----
    if (wave < 6) {
        const int cen = sCen[wave];
        const bool validk = (sVal[wave] != NINF);
        int start = cen - 10;
        start = start < 0 ? 0 : (start > LNUM - WINN ? LNUM - WINN : start);
        const int i = lane;
        const bool on = (i < WINN);
        float lam_i = lam[start + (on ? i : 0)];
        float yt = on ? tPad[start + i + 6] : 0.0f;
        float yp = on ? prow[start + i] : 0.0f;
        float l0  = __shfl(lam_i, 0, 32), l1  = __shfl(lam_i, 20, 32);
        float yt0 = __shfl(yt, 0, 32),    yt1 = __shfl(yt, 20, 32);
        float yp0 = __shfl(yp, 0, 32),    yp1 = __shfl(yp, 20, 32);
        float tpar  = (lam_i - l0) / (l1 - l0 + 1e-6f);
        float contT = fmaxf((1.0f - tpar) * yt0 + tpar * yt1, 1e-6f);
        float contP = fmaxf((1.0f - tpar) * yp0 + tpar * yp1, 1e-6f);
        float dt = fminf(fmaxf(1.0f - fminf(fmaxf(yt / contT, 0.0f), 2.0f), 0.0f), 1.0f);
        float dp = fminf(fmaxf(1.0f - fminf(fmaxf(yp / contP, 0.0f), 2.0f), 0.0f), 1.0f);
        if (!on) { dt = 0.0f; dp = 0.0f; }
        float dtn = __shfl_down(dt, 1, 32);
        float dpn = __shfl_down(dp, 1, 32);
        bool inter = (i < WINN - 1);
        float aT  = inter ? 0.25f * (dt + dtn) : 0.0f;   // 0.5*(d[i]+d[i+1])*dl, dl = 0.5 exact
        float aP  = inter ? 0.25f * (dp + dpn) : 0.0f;
        float wt  = on ? dt + 1e-7f : 0.0f;
        float wp  = on ? dp + 1e-7f : 0.0f;
        float wtl = on ? lam_i * wt : 0.0f;
        float wpl = on ? lam_i * wp : 0.0f;
        float ad  = on ? fabsf(dp - dt) : 0.0f;
#pragma unroll
        for (int o = 16; o > 0; o >>= 1) {               // wave32 shuffle reductions
            aT  += __shfl_down(aT,  o, 32);
            aP  += __shfl_down(aP,  o, 32);
            wt  += __shfl_down(wt,  o, 32);
            wp  += __shfl_down(wp,  o, 32);
            wtl += __shfl_down(wtl, o, 32);
            wpl += __shfl_down(wpl, o, 32);
            ad  += __shfl_down(ad,  o, 32);
        }
        if (lane == 0 && validk && aT >= 1e-5f) {
            float relE = fabsf(aP - aT) / (aT + 1e-7f);
            float uf   = fmaxf(aT - aP, 0.0f) / (aT + 1e-7f);
            float areaTerm = relE + uf;                  // (UNDERFILL-1)*uf, UNDERFILL=2
            float cenTerm  = fabsf(wpl / wp - wtl / wt);
            float depTerm  = ad * (1.0f / 21.0f);
            atomicAdd(&lacc[0], areaTerm);
            atomicAdd(&lacc[1], cenTerm);
            atomicAdd(&lacc[2], depTerm);
            atomicAdd(&lacc[3], 1.0f);
        }
    }
    __syncthreads();
    if (tid < 4) atomicAdd(&gacc[tid], lacc[tid]);
}

__global__ void dip_init(float* gacc) { if (threadIdx.x < 4) gacc[threadIdx.x] = 0.0f; }

__global__ void dip_final(const float* __restrict__ gacc, float* __restrict__ out) {
    float n = gacc[3];
    // W_AREA=1, W_CENTROID=1, W_DEPTH=0.2, shared 1/max(n,1)
    out[0] = (n > 0.0f) ? (gacc[0] + gacc[1] + 0.2f * gacc[2]) / fmaxf(n, 1.0f) : 0.0f;
}

extern "C" void kernel_launch(void* const* d_in, const int* in_sizes, int n_in,
                              void* d_out, int out_size, void* d_ws, size_t ws_size,
                              hipStream_t stream) {
    const float* pred = (const float*)d_in[0];
    const float* tgt  = (const float*)d_in[1];
    const float* lam  = (const float*)d_in[2];
    float* out  = (float*)d_out;
    float* gacc = (float*)d_ws;
    const int B = in_sizes[0] / LNUM;
    dip_init<<<1, 4, 0, stream>>>(gacc);
    dip_main<<<B, 256, 0, stream>>>(pred, tgt, lam, gacc);
    dip_final<<<1, 1, 0, stream>>>(gacc, out);
}